// SMPLDeformer_39788577030668
// MI455X (gfx1250) — compile-verified
//
#include <hip/hip_runtime.h>
#include <hip/hip_bf16.h>

// ---------------------------------------------------------------------------
// SMPL deformer for MI455X (gfx1250, wave32).
//  k0: tfs = tfs_A @ tfs_inv, zero-padded J 55->56            -> ws
//  k1: off_full[b,m,3] = betas@spdir + pose_feature@podir     -> ws
//  k2: fused per-point kernel:
//      gather face offsets, canonical x, trilinear/masked weights,
//      normalize, T = w x tfs via V_WMMA_F32_16X16X4_F32 (16 pts/wave,
//      14 K-steps), coalesced T writeback, posed = T @ xh.
//      Voxel grid (55 MB) is L2-resident on the 192 MB L2.
// ---------------------------------------------------------------------------

constexpr int kB  = 4;
constexpr int kN  = 131072;
constexpr int kM  = 10475;
constexpr int kJ  = 55;
constexpr int kD  = 64;
constexpr int kPF = 486;
constexpr int kKP = 56;            // J padded to multiple of 4 for WMMA K
constexpr int kWS = 68;            // w_lds row stride (floats): 272B keeps 8B
                                   // alignment and gives bank = (4m+k)%64 ->
                                   // conflict-free A-fragment loads
constexpr int kWavesPerBlock = 8;  // 256 threads
constexpr int kPtsPerWave    = 16;
constexpr int kPtsPerBlock   = kWavesPerBlock * kPtsPerWave;  // 128

typedef __attribute__((ext_vector_type(2))) float v2f;
typedef __attribute__((ext_vector_type(8))) float v8f;

// ---------------- kernel 0: per-joint 4x4 transform products ----------------
__global__ __launch_bounds__(256) void tfs_prep_kernel(
    const float* __restrict__ tfs_A,    // [B,J,4,4]
    const float* __restrict__ tfs_inv,  // [1,J,4,4]
    float* __restrict__ tfs_ws)         // [B,kKP,16] (rows J..kKP-1 zero)
{
    int t = blockIdx.x * blockDim.x + threadIdx.x;
    if (t >= kB * kKP * 16) return;
    int rc = t & 15;
    int j  = (t >> 4) % kKP;
    int b  = t / (kKP * 16);
    float v = 0.f;
    if (j < kJ) {
        int r = rc >> 2, c = rc & 3;
        const float* A  = tfs_A  + ((size_t)b * kJ + j) * 16;
        const float* Iv = tfs_inv + (size_t)j * 16;
#pragma unroll
        for (int k = 0; k < 4; ++k) v += A[r * 4 + k] * Iv[k * 4 + c];
    }
    tfs_ws[t] = v;
}

// ------------- kernel 1: per-template-vertex blendshape offsets -------------
__global__ __launch_bounds__(256) void offsets_prep_kernel(
    const float* __restrict__ betas,   // [B,10]
    const float* __restrict__ posef,   // [B,PF]
    const float* __restrict__ spdir,   // [M,3,10]
    const float* __restrict__ podir,   // [PF,3*M]
    float* __restrict__ off_full)      // [B,M,3]
{
    int col = blockIdx.x * blockDim.x + threadIdx.x;  // 0..3M-1
    if (col >= kM * 3) return;
    float acc[kB] = {0.f, 0.f, 0.f, 0.f};
    // pose_full: read each podir column element once, FMA into all 4 batches
    for (int l = 0; l < kPF; ++l) {
        float pv = podir[(size_t)l * (kM * 3) + col];
#pragma unroll
        for (int b = 0; b < kB; ++b) acc[b] += posef[b * kPF + l] * pv;
    }
    // shape_full: spdir element (m,k,l) at col*10+l since col == m*3+k
#pragma unroll
    for (int l = 0; l < 10; ++l) {
        float sv = spdir[(size_t)col * 10 + l];
#pragma unroll
        for (int b = 0; b < kB; ++b) acc[b] += betas[b * 10 + l] * sv;
    }
#pragma unroll
    for (int b = 0; b < kB; ++b) off_full[(size_t)b * kM * 3 + col] = acc[b];
}

// ---------------------- kernel 2: fused per-point path ----------------------
__global__ __launch_bounds__(256) void deform_main_kernel(
    const float* __restrict__ pts,       // [B,N,3]
    const int*   __restrict__ faces,     // [N,3]
    const float* __restrict__ off_full,  // [B,M,3]
    const float* __restrict__ poseoff,   // [N,3]
    const float* __restrict__ lbsw,      // [N,J]
    const float* __restrict__ vox,       // [J,D,D,D]
    const float* __restrict__ vscale,    // [3]
    const float* __restrict__ voffs,     // [3]
    const int*   __restrict__ mask,      // [B,N]
    const float* __restrict__ tfs_ws,    // [B,kKP,16]
    float* __restrict__ out_posed,       // [B,N,3]
    float* __restrict__ out_T)           // [B,N,16]
{
    __shared__ float tfs_lds[kKP][16];                                   // 3.5 KB
    __shared__ float w_lds[kWavesPerBlock][16][kWS];                     // 34 KB
    __shared__ float x_lds[kWavesPerBlock][16][4];
    __shared__ float cw_lds[kWavesPerBlock][16][8];
    __shared__ int   cf_lds[kWavesPerBlock][16][8];
    __shared__ float scale_lds[kWavesPerBlock][16];
    __shared__ int   mask_lds[kWavesPerBlock][16];
    __shared__ alignas(16) float T_lds[kWavesPerBlock][16][16];          // 8 KB

    const int b    = blockIdx.y;
    const int tid  = threadIdx.x;
    const int wave = tid >> 5;
    const int lane = tid & 31;
    const int half = lane >> 4;       // 0 or 1 (lane half, per WMMA f32 layouts)
    const int m    = lane & 15;       // point-in-wave / column index
    const int p_base = blockIdx.x * kPtsPerBlock + wave * kPtsPerWave;

    // stage tfs[b] (padded 56x16) into LDS cooperatively
    for (int i = tid; i < kKP * 16; i += 256)
        (&tfs_lds[0][0])[i] = tfs_ws[(size_t)b * kKP * 16 + i];

    // ---- Phase A: per-point setup (lanes 0..15 of each wave) ----
    if (half == 0) {
        int gp = p_base + m;
        int f0 = faces[gp * 3 + 0], f1 = faces[gp * 3 + 1], f2 = faces[gp * 3 + 2];
        const float* ob = off_full + (size_t)b * kM * 3;
        const float third = 1.f / 3.f;
        float px = pts[((size_t)b * kN + gp) * 3 + 0];
        float py = pts[((size_t)b * kN + gp) * 3 + 1];
        float pz = pts[((size_t)b * kN + gp) * 3 + 2];
        float ox = (ob[f0 * 3 + 0] + ob[f1 * 3 + 0] + ob[f2 * 3 + 0]) * third;
        float oy = (ob[f0 * 3 + 1] + ob[f1 * 3 + 1] + ob[f2 * 3 + 1]) * third;
        float oz = (ob[f0 * 3 + 2] + ob[f1 * 3 + 2] + ob[f2 * 3 + 2]) * third;
        x_lds[wave][m][0] = px + ox - poseoff[gp * 3 + 0];
        x_lds[wave][m][1] = py + oy - poseoff[gp * 3 + 1];
        x_lds[wave][m][2] = pz + oz - poseoff[gp * 3 + 2];
        x_lds[wave][m][3] = 1.f;

        // trilinear setup on the *raw* pts (matches reference)
        float gx = fminf(fmaxf(px * vscale[0] + voffs[0], -1.f), 1.f);
        float gy = fminf(fmaxf(py * vscale[1] + voffs[1], -1.f), 1.f);
        float gz = fminf(fmaxf(pz * vscale[2] + voffs[2], -1.f), 1.f);
        float tx = (gx + 1.f) * 0.5f * (float)(kD - 1);
        float ty = (gy + 1.f) * 0.5f * (float)(kD - 1);
        float tz = (gz + 1.f) * 0.5f * (float)(kD - 1);
        float fix = fminf(fmaxf(floorf(tx), 0.f), (float)(kD - 2));
        float fiy = fminf(fmaxf(floorf(ty), 0.f), (float)(kD - 2));
        float fiz = fminf(fmaxf(floorf(tz), 0.f), (float)(kD - 2));
        int ix = (int)fix, iy = (int)fiy, iz = (int)fiz;
        float fx = tx - fix, fy = ty - fiy, fz = tz - fiz;
#pragma unroll
        for (int dx = 0; dx < 2; ++dx)
#pragma unroll
            for (int dy = 0; dy < 2; ++dy)
#pragma unroll
                for (int dz = 0; dz < 2; ++dz) {
                    int c = dx * 4 + dy * 2 + dz;
                    float wgt = (dx ? fx : 1.f - fx) *
                                (dy ? fy : 1.f - fy) *
                                (dz ? fz : 1.f - fz);
                    cw_lds[wave][m][c] = wgt;
                    cf_lds[wave][m][c] = ((ix + dx) * kD + (iy + dy)) * kD + (iz + dz);
                }
        mask_lds[wave][m] = mask[(size_t)b * kN + gp];
    }
    __syncthreads();

    // ---- Phase B: fill skinning weights w[16][56] (all 32 lanes) ----
    {
        int gp = p_base + m;
        int mk = mask_lds[wave][m];
#pragma unroll 4
        for (int it = 0; it < kKP / 2; ++it) {
            int j = 2 * it + half;
            float wv = 0.f;
            if (j < kJ) {
                if (mk > 0) {
                    wv = lbsw[(size_t)gp * kJ + j];
                } else {
                    const float* vj = vox + (size_t)j * (kD * kD * kD);
                    float s = 0.f;
#pragma unroll
                    for (int c = 0; c < 8; ++c)
                        s += vj[cf_lds[wave][m][c]] * cw_lds[wave][m][c];
                    wv = s;
                }
            }
            w_lds[wave][m][j] = wv;
        }
    }
    __syncthreads();

    // ---- Phase C: per-point normalization scale ----
    if (half == 0) {
        float s = 0.f;
#pragma unroll
        for (int j = 0; j < kKP; ++j) s += w_lds[wave][m][j];
        scale_lds[wave][m] = 1.f / fmaxf(s, 1e-8f);
    }
    __syncthreads();

    // ---- Phase D: T = w x tfs via V_WMMA_F32_16X16X4_F32, K = 56 in 14 steps
    // A (16x4 f32): lane half 0 -> K={k0,k0+1}, half 1 -> K={k0+2,k0+3}, row M=m
    // B (4x16 f32): same half-K split, column N=m
    // D (16x16 f32): lane holds column N=m of rows M = r + 8*half, r=0..7
    v8f acc = {};
    float sc = scale_lds[wave][m];
#pragma unroll
    for (int ks = 0; ks < kKP / 4; ++ks) {
        int k0 = 4 * ks + 2 * half;
        v2f av, bv;
        av.x = w_lds[wave][m][k0]     * sc;
        av.y = w_lds[wave][m][k0 + 1] * sc;
        bv.x = tfs_lds[k0][m];
        bv.y = tfs_lds[k0 + 1][m];
        acc = __builtin_amdgcn_wmma_f32_16x16x4_f32(
            /*neg_a=*/false, av, /*neg_b=*/false, bv,
            /*c_mod=*/(short)0, acc, /*reuse_a=*/false, /*reuse_b=*/false);
    }

    // ---- Phase E: stage T through LDS (scatter by lane layout) ----
#pragma unroll
    for (int r = 0; r < 8; ++r)
        T_lds[wave][r + 8 * half][m] = acc[r];
    __syncthreads();

    // ---- Phase E': coalesced T writeback (wave's 16 pts = 1KB contiguous) ----
    {
        const float4* tsrc = (const float4*)&T_lds[wave][0][0];  // 64 float4
        float4* tdst = (float4*)(out_T + ((size_t)b * kN + p_base) * 16);
#pragma unroll
        for (int q = 0; q < 2; ++q) {
            int idx = lane + 32 * q;
            tdst[idx] = tsrc[idx];
        }
    }

    // ---- Phase F: posed = (T @ xh)[:3], one b96 store per point ----
    if (half == 0) {
        float x0 = x_lds[wave][m][0];
        float x1 = x_lds[wave][m][1];
        float x2 = x_lds[wave][m][2];
        float3 pr;
        pr.x = T_lds[wave][m][0] * x0 + T_lds[wave][m][1] * x1 +
               T_lds[wave][m][2] * x2 + T_lds[wave][m][3];
        pr.y = T_lds[wave][m][4] * x0 + T_lds[wave][m][5] * x1 +
               T_lds[wave][m][6] * x2 + T_lds[wave][m][7];
        pr.z = T_lds[wave][m][8] * x0 + T_lds[wave][m][9] * x1 +
               T_lds[wave][m][10] * x2 + T_lds[wave][m][11];
        ((float3*)out_posed)[(size_t)b * kN + p_base + m] = pr;
    }
}

// ---------------------------------------------------------------------------
extern "C" void kernel_launch(void* const* d_in, const int* in_sizes, int n_in,
                              void* d_out, int out_size, void* d_ws, size_t ws_size,
                              hipStream_t stream) {
    (void)in_sizes; (void)n_in; (void)out_size; (void)ws_size;

    const float* pts      = (const float*)d_in[0];
    const float* betas    = (const float*)d_in[1];
    const float* posef    = (const float*)d_in[2];
    const float* spdir    = (const float*)d_in[3];
    const float* podir    = (const float*)d_in[4];
    const int*   faces    = (const int*)  d_in[5];
    const float* tfs_A    = (const float*)d_in[6];
    const float* tfs_inv  = (const float*)d_in[7];
    const float* poseoff  = (const float*)d_in[8];
    const float* lbsw     = (const float*)d_in[9];
    const float* vox      = (const float*)d_in[10];
    const float* vscale   = (const float*)d_in[11];
    const float* voffs    = (const float*)d_in[12];
    const int*   mask     = (const int*)  d_in[13];

    float* out_posed = (float*)d_out;                          // [B,N,1,3]
    float* out_T     = out_posed + (size_t)kB * kN * 3;        // [B,N,4,4]

    float* tfs_ws   = (float*)d_ws;                            // [B,kKP,16]
    float* off_full = tfs_ws + (size_t)kB * kKP * 16;          // [B,M,3]

    // k0: tfs products (B*kKP*16 = 3584 elems)
    tfs_prep_kernel<<<(kB * kKP * 16 + 255) / 256, 256, 0, stream>>>(
        tfs_A, tfs_inv, tfs_ws);

    // k1: blendshape offsets (3M columns)
    offsets_prep_kernel<<<(kM * 3 + 255) / 256, 256, 0, stream>>>(
        betas, posef, spdir, podir, off_full);

    // k2: fused per-point deform, 128 points per block
    dim3 grid(kN / kPtsPerBlock, kB);
    deform_main_kernel<<<grid, 256, 0, stream>>>(
        pts, faces, off_full, poseoff, lbsw, vox, vscale, voffs, mask,
        tfs_ws, out_posed, out_T);
}